// CtxAttnGuidedMask_63453846831115
// MI455X (gfx1250) — compile-verified
//
#include <hip/hip_runtime.h>
#include <hip/hip_bf16.h>

typedef float v2f __attribute__((ext_vector_type(2)));
typedef float v8f __attribute__((ext_vector_type(8)));

#define B_ 32
#define N_ 4096
#define D_ 512
#define EPS_ 1e-6f

// ---------------------------------------------------------------------------
// Kernel 1: L2-normalize cond_feat per batch. grid=32 blocks x 32 threads.
// ---------------------------------------------------------------------------
__global__ void cgm_norm_cond(const float* __restrict__ cond,
                              float* __restrict__ cond_n) {
    int b = blockIdx.x;
    int l = threadIdx.x;                    // 0..31 (wave32)
    const float* src = cond + (size_t)b * D_;
    float* dst = cond_n + (size_t)b * D_;

    float ss = 0.0f;
    float4 v[4];
#pragma unroll
    for (int j = 0; j < 4; ++j) {
        v[j] = *(const float4*)(src + l * 16 + j * 4);
        ss = fmaf(v[j].x, v[j].x, ss);
        ss = fmaf(v[j].y, v[j].y, ss);
        ss = fmaf(v[j].z, v[j].z, ss);
        ss = fmaf(v[j].w, v[j].w, ss);
    }
#pragma unroll
    for (int off = 16; off >= 1; off >>= 1)
        ss += __shfl_xor(ss, off, 32);
    float inv = 1.0f / fmaxf(sqrtf(ss), EPS_);
#pragma unroll
    for (int j = 0; j < 4; ++j) {
        float4 o = v[j];
        o.x *= inv; o.y *= inv; o.z *= inv; o.w *= inv;
        *(float4*)(dst + l * 16 + j * 4) = o;
    }
}

// ---------------------------------------------------------------------------
// Kernel 2: cosine scores via V_WMMA_F32_16X16X4_F32.
// One wave per 16-token tile; 128 K-steps of K=4.
// A (16x4 f32): lane l holds row (l&15), float2 at K offset 2*(l>>4).
// B (4x16 f32): all 16 columns = cond chunk -> same per-lane float2 mapping.
// D (16x16 f32): all columns equal dot(token_m, cond_n).
// grid = 1024 blocks x 256 threads (8 waves/block), tiles = 32*4096/16 = 8192.
// ---------------------------------------------------------------------------
__global__ void cgm_scores_wmma(const float* __restrict__ ctx,
                                const float* __restrict__ cond_n,
                                float* __restrict__ scores) {
    int wave = (blockIdx.x << 3) + (threadIdx.x >> 5);   // tile id, 0..8191
    int b    = wave >> 8;                                // 256 tiles per batch
    int trow = (wave & 255) << 4;                        // first token of tile
    int l    = threadIdx.x & 31;
    int row  = l & 15;
    int hi   = l >> 4;

    const float* rowp = ctx + ((size_t)(b * N_ + trow + row)) * D_ + 2 * hi;
    const float* cndp = cond_n + (size_t)b * D_ + 2 * hi;

    v8f acc = {0.f, 0.f, 0.f, 0.f, 0.f, 0.f, 0.f, 0.f};
    float ss = 0.0f;

#pragma unroll 4
    for (int k0 = 0; k0 < D_; k0 += 4) {
        float2 a  = *(const float2*)(rowp + k0);
        float2 bb = *(const float2*)(cndp + k0);
        v2f va = {a.x, a.y};
        v2f vb = {bb.x, bb.y};
        acc = __builtin_amdgcn_wmma_f32_16x16x4_f32(
            false, va, false, vb, (short)0, acc, false, false);
        ss = fmaf(a.x, a.x, ss);
        ss = fmaf(a.y, a.y, ss);
    }

    // full-row sumsq: lane q and q^16 cover complementary K halves of row q&15
    float ssFull  = ss + __shfl_xor(ss, 16, 32);   // lane q: sumsq(token q&15)
    float ssOther = __shfl_xor(ssFull, 8, 32);

    // writer lanes: 0..7 -> token sel (M=v,hi=0); 16..23 -> token 8+sel (M=v+8)
    int sel = l & 7;
    if ((l & 8) == 0) {
        int t = sel + (hi << 3);
        float dot = acc[sel];
        float ssv = hi ? ssOther : ssFull;   // hi=1: lane l^8 holds sumsq(l-8)=sumsq(t)
        float score = dot / fmaxf(sqrtf(ssv), EPS_);
        scores[(size_t)b * N_ + trow + t] = score;
    }
}

// ---------------------------------------------------------------------------
// Kernel 3: per-batch exact top-k selection with lowest-index tie-break.
// One workgroup (256 threads) per batch. Keys in LDS, MSB-first bit search.
// ---------------------------------------------------------------------------
__global__ void cgm_select(const float* __restrict__ scores,
                           const int* __restrict__ kptr,
                           int* __restrict__ flags) {
    __shared__ unsigned keys[N_];
    __shared__ int s_cnt;
    __shared__ int s_pref[256];
    __shared__ int s_need;

    int b = blockIdx.x;
    int tid = threadIdx.x;
    int k = *kptr;

    // load + monotonic float->uint key (descending order == uint descending)
    for (int n = tid; n < N_; n += 256) {
        unsigned u = __float_as_uint(scores[(size_t)b * N_ + n]);
        keys[n] = (u & 0x80000000u) ? ~u : (u | 0x80000000u);
    }
    __syncthreads();

    // find kv = k-th largest key: max v with count(keys >= v) >= k
    unsigned kv = 0;
    for (int bit = 31; bit >= 0; --bit) {
        unsigned cand = kv | (1u << bit);
        if (tid == 0) s_cnt = 0;
        __syncthreads();
        int local = 0;
        for (int n = tid; n < N_; n += 256)
            local += (keys[n] >= cand) ? 1 : 0;
        atomicAdd(&s_cnt, local);
        __syncthreads();
        if (s_cnt >= k) kv = cand;
        __syncthreads();
    }

    // count strictly-greater, derive how many ties to keep
    if (tid == 0) s_cnt = 0;
    __syncthreads();
    {
        int local = 0;
        for (int n = tid; n < N_; n += 256)
            local += (keys[n] > kv) ? 1 : 0;
        atomicAdd(&s_cnt, local);
    }
    __syncthreads();
    if (tid == 0) s_need = k - s_cnt;
    __syncthreads();
    int need_eq = s_need;

    // exclusive prefix (by index) of (key == kv): contiguous 16-elem chunks
    int base = tid * 16;
    int local_eq = 0;
#pragma unroll
    for (int j = 0; j < 16; ++j)
        local_eq += (keys[base + j] == kv) ? 1 : 0;
    s_pref[tid] = local_eq;
    __syncthreads();
    if (tid == 0) {
        int run = 0;
        for (int t = 0; t < 256; ++t) {
            int c = s_pref[t];
            s_pref[t] = run;
            run += c;
        }
    }
    __syncthreads();

    int rank = s_pref[tid];
#pragma unroll
    for (int j = 0; j < 16; ++j) {
        unsigned key = keys[base + j];
        bool eq = (key == kv);
        bool selg = (key > kv);
        int selected = (selg || (eq && rank < need_eq)) ? 1 : 0;
        if (eq) rank++;
        flags[(size_t)b * N_ + base + j] = selected;
    }
}

// ---------------------------------------------------------------------------
// Kernel 4: write output. float4/thread; flag is uniform across the 4 waves
// of a token (128 float4 lanes/token), so the masked branch skips ctx reads.
// grid = 65536 x 256.
// ---------------------------------------------------------------------------
__global__ void cgm_apply(const float4* __restrict__ ctx4,
                          const float* __restrict__ mask_tok,
                          const int* __restrict__ flags,
                          float4* __restrict__ out4) {
    size_t i = (size_t)blockIdx.x * blockDim.x + threadIdx.x; // float4 index
    int token = (int)(i >> 7);           // 512/4 = 128 float4 per token
    int d4    = (int)(i & 127);
    if (flags[token]) {
        out4[i] = ((const float4*)mask_tok)[d4];
    } else {
        out4[i] = ctx4[i];
    }
}

// ---------------------------------------------------------------------------
extern "C" void kernel_launch(void* const* d_in, const int* in_sizes, int n_in,
                              void* d_out, int out_size, void* d_ws, size_t ws_size,
                              hipStream_t stream) {
    const float* ctx  = (const float*)d_in[0];   // [32,4096,512] f32
    const float* cond = (const float*)d_in[1];   // [32,512] f32
    const float* mtok = (const float*)d_in[2];   // [512] f32
    const int*   kptr = (const int*)d_in[3];     // scalar k on device
    float* out = (float*)d_out;

    char* ws = (char*)d_ws;
    float* cond_n = (float*)ws;                                   // 64 KB
    float* scores = (float*)(ws + (size_t)B_ * D_ * 4);           // 512 KB
    int*   flags  = (int*)(ws + (size_t)B_ * D_ * 4
                              + (size_t)B_ * N_ * 4);             // 512 KB

    cgm_norm_cond<<<B_, 32, 0, stream>>>(cond, cond_n);
    cgm_scores_wmma<<<(B_ * N_ / 16) / 8, 256, 0, stream>>>(ctx, cond_n, scores);
    cgm_select<<<B_, 256, 0, stream>>>(scores, kptr, flags);
    cgm_apply<<<(int)(((size_t)B_ * N_ * D_ / 4) / 256), 256, 0, stream>>>(
        (const float4*)ctx, mtok, flags, (float4*)out);
}